// DEHNN_layer_15590731285069
// MI455X (gfx1250) — compile-verified
//
#include <hip/hip_runtime.h>

typedef __attribute__((ext_vector_type(2))) float v2f;
typedef __attribute__((ext_vector_type(8))) float v8f;
typedef __attribute__((ext_vector_type(4))) float f4;

#define NROWS 8192
#define DF    256
#define NEDGE 131072

#define BM    64
#define BK    32
#define CT_N  8      // 16-col tiles per wave (covers 128 cols)
#define LDA_S 36     // padded LDS row stride for A tile (floats); mult of 4 for b128 stores
#define PLDB  544    // dword stride per B k-pair row: even (b64 align), %64==32 (bank split)

// -------------------------------------------------------------------------
// scale[row] = 1 / (sum_j A[row][j] + 1)
__global__ __launch_bounds__(256)
void rowscale_kernel(const float* __restrict__ A, float* __restrict__ scale) {
    __shared__ float red[256];
    const int row = blockIdx.x;
    const f4* rp4 = (const f4*)(A + (size_t)row * NROWS);
    float s = 0.f;
    for (int c = threadIdx.x; c < NROWS / 4; c += 256) {
        f4 v = rp4[c];
        s += (v.x + v.y) + (v.z + v.w);
    }
    red[threadIdx.x] = s;
    __syncthreads();
    for (int off = 128; off > 0; off >>= 1) {
        if (threadIdx.x < off) red[threadIdx.x] += red[threadIdx.x + off];
        __syncthreads();
    }
    if (threadIdx.x == 0) scale[row] = 1.0f / (red[0] + 1.0f);
}

// -------------------------------------------------------------------------
// C = A[M x K] @ B[K x 256] via V_WMMA_F32_16X16X4_F32.
// Block: 256 threads = 8 waves laid out 4 row-groups x 2 column-halves.
// Each wave: 16 rows x 128 cols = 8 v8f accumulators.
// Double-buffered LDS, one barrier per BK chunk. B stored k-pair interleaved
// so each B fragment is a single aligned ds_load_b64 (no repack movs).
// Fragment loads are gathered into register arrays ahead of the WMMA bursts
// so the scheduler can clause the DS loads and use partial dscnt waits.
// MODE 0: out = (addin + C) * scale[row]       (bottom-to-up level)
// MODE 1: out = C                              (h = x @ W)
template <int MODE>
__global__ __launch_bounds__(256)
void gemm_wmma_f32(const float* __restrict__ A, const float* __restrict__ B,
                   const float* __restrict__ addin, const float* __restrict__ scale,
                   float* __restrict__ out, int K) {
    __shared__ float sA[2][BM * LDA_S];            // 2 x 9216 B
    __shared__ float sB[2][(BK / 2) * PLDB];       // 2 x 34816 B

    const int tid     = threadIdx.x;
    const int wave    = tid >> 5;
    const int lane    = tid & 31;
    const int lhalf   = lane >> 4;     // 0 or 1
    const int l16     = lane & 15;
    const int rowGrp  = wave >> 1;     // 0..3  -> rows rowGrp*16 .. +15
    const int colHalf = wave & 1;      // 0..1  -> cols colHalf*128 .. +127

    const int rowBase = blockIdx.x * BM;

    v8f acc[CT_N];
#pragma unroll
    for (int t = 0; t < CT_N; ++t) {
        v8f z = {0.f, 0.f, 0.f, 0.f, 0.f, 0.f, 0.f, 0.f};
        acc[t] = z;
    }

    f4 ra[2];   // A-tile prefetch registers (BM x BK = 512 float4 / 256 thr)
    f4 rb[8];   // B-tile prefetch registers (BK x 256 = 2048 float4 / 256 thr)

    auto loadG = [&](int k0) {
#pragma unroll
        for (int i = 0; i < 2; ++i) {
            int li = tid + i * 256;
            int r  = li >> 3;            // 8 float4 per A row
            int c4 = (li & 7) << 2;
            ra[i] = *(const f4*)(A + (size_t)(rowBase + r) * K + k0 + c4);
        }
#pragma unroll
        for (int i = 0; i < 8; ++i) {
            int li = tid + i * 256;
            int r  = li >> 6;            // 64 float4 per B row
            int c4 = (li & 63) << 2;
            rb[i] = *(const f4*)(B + (size_t)(k0 + r) * DF + c4);
        }
    };

    auto storeL = [&](int buf) {
#pragma unroll
        for (int i = 0; i < 2; ++i) {
            int li = tid + i * 256;
            int r  = li >> 3;
            int c4 = (li & 7) << 2;
            *(f4*)&sA[buf][r * LDA_S + c4] = ra[i];
        }
#pragma unroll
        for (int i = 0; i < 8; ++i) {
            int li = tid + i * 256;
            int r  = li >> 6;
            int c4 = (li & 63) << 2;
            float* dst = &sB[buf][(r >> 1) * PLDB + (r & 1)];
            dst[2 * (c4 + 0)] = rb[i].x;
            dst[2 * (c4 + 1)] = rb[i].y;
            dst[2 * (c4 + 2)] = rb[i].z;
            dst[2 * (c4 + 3)] = rb[i].w;
        }
    };

    auto compute = [&](int buf) {
        const float* pA = sA[buf];
        const float* pB = sB[buf];
        // gather the whole chunk's A fragments up front (8 x v2f)
        v2f afr[BK / 4];
#pragma unroll
        for (int ks = 0; ks < BK / 4; ++ks)
            afr[ks] = *(const v2f*)&pA[(rowGrp * 16 + l16) * LDA_S + ks * 4 + 2 * lhalf];
#pragma unroll
        for (int ks = 0; ks < BK / 4; ++ks) {
            const int p = ks * 2 + lhalf;                 // B k-pair row
            // gather this k-step's 8 B fragments, then burst 8 WMMAs
            v2f bfr[CT_N];
#pragma unroll
            for (int ct = 0; ct < CT_N; ++ct) {
                const int col = colHalf * 128 + ct * 16 + l16;
                bfr[ct] = *(const v2f*)&pB[p * PLDB + 2 * col];
            }
#pragma unroll
            for (int ct = 0; ct < CT_N; ++ct) {
                acc[ct] = __builtin_amdgcn_wmma_f32_16x16x4_f32(
                    /*neg_a=*/false, afr[ks], /*neg_b=*/false, bfr[ct],
                    /*c_mod=*/(short)0, acc[ct],
                    /*reuse_a=*/false, /*reuse_b=*/false);
            }
        }
    };

    // pipeline: prefetch chunk c+1 from global while computing chunk c
    loadG(0);
    storeL(0);
    __syncthreads();
    const int nc = K / BK;
    for (int c = 0; c < nc; ++c) {
        const int cur = c & 1;
        if (c + 1 < nc) loadG((c + 1) * BK);
        compute(cur);
        if (c + 1 < nc) storeL(cur ^ 1);
        __syncthreads();
    }

    // epilogue: C layout — VGPR v : M = v + 8*lhalf, N = lane&15
    const int r0 = rowBase + rowGrp * 16 + 8 * lhalf;
#pragma unroll
    for (int v = 0; v < 8; ++v) {
        const int row = r0 + v;
        float s = 1.0f;
        if (MODE == 0) s = scale[row];
#pragma unroll
        for (int ct = 0; ct < CT_N; ++ct) {
            const int col = colHalf * 128 + ct * 16 + l16;
            const size_t idx = (size_t)row * DF + col;
            float c = acc[ct][v];
            if (MODE == 0) out[idx] = (addin[idx] + c) * s;
            else           out[idx] = c;
        }
    }
}

// -------------------------------------------------------------------------
__global__ __launch_bounds__(256)
void deg_init_kernel(float* __restrict__ deg) {
    deg[blockIdx.x * 256 + threadIdx.x] = 1.0f;   // self-loop
}

__global__ __launch_bounds__(256)
void deg_count_kernel(const int* __restrict__ dst, float* __restrict__ deg) {
    int e = blockIdx.x * 256 + threadIdx.x;
    if (e < NEDGE) atomicAdd(&deg[dst[e]], 1.0f);
}

__global__ __launch_bounds__(256)
void deg_finish_kernel(float* __restrict__ deg) {
    int i = blockIdx.x * 256 + threadIdx.x;
    deg[i] = rsqrtf(deg[i]);                      // deg -> dinv
}

// out[i][d] = dinv[i]^2 * h[i][d] + bias[d]   (self-loop + bias init)
__global__ __launch_bounds__(256)
void selfloop_bias_kernel(const float* __restrict__ h, const float* __restrict__ dinv,
                          const float* __restrict__ bias, float* __restrict__ out) {
    int i = blockIdx.x * 256 + threadIdx.x;       // over N*DF
    int row = i >> 8;
    int col = i & 255;
    float di = dinv[row];
    out[i] = di * di * h[i] + bias[col];
}

// out[dst] += dinv[src]*dinv[dst] * h[src]  per edge (atomic scatter)
// 256 threads handle 4 edges (64 threads x 4 floats per edge).
__global__ __launch_bounds__(256)
void scatter_edges_kernel(const int* __restrict__ src, const int* __restrict__ dst,
                          const float* __restrict__ dinv, const float* __restrict__ h,
                          float* __restrict__ out) {
    const int e = blockIdx.x * 4 + (threadIdx.x >> 6);
    const int s = src[e];
    const int d = dst[e];
    const float nrm = dinv[s] * dinv[d];
    const int c = (threadIdx.x & 63) * 4;
    f4 hv = *(const f4*)(h + (size_t)s * DF + c);
    float* op = out + (size_t)d * DF + c;
    atomicAdd(op + 0, nrm * hv.x);
    atomicAdd(op + 1, nrm * hv.y);
    atomicAdd(op + 2, nrm * hv.z);
    atomicAdd(op + 3, nrm * hv.w);
}

// -------------------------------------------------------------------------
extern "C" void kernel_launch(void* const* d_in, const int* in_sizes, int n_in,
                              void* d_out, int out_size, void* d_ws, size_t ws_size,
                              hipStream_t stream) {
    const float* embedding = (const float*)d_in[0];           // [N, 256]
    const float* d2u       = (const float*)d_in[1];           // [2, N, N]
    const int*   same_ei   = (const int*)d_in[2];             // [2, E]
    const int*   u2d_ei    = (const int*)d_in[3];             // [2, E]
    const float* W_same    = (const float*)d_in[4];           // [256, 256]
    const float* b_same    = (const float*)d_in[5];           // [256]
    const float* W_top     = (const float*)d_in[6];           // [256, 256]
    const float* b_top     = (const float*)d_in[7];           // [256]
    float*       out       = (float*)d_out;                   // [N, 256]

    float* buf0  = (float*)d_ws;                              // N*DF
    float* buf1  = buf0 + (size_t)NROWS * DF;                 // N*DF
    float* bufh  = buf1 + (size_t)NROWS * DF;                 // N*DF
    float* scale = bufh + (size_t)NROWS * DF;                 // N
    float* dinv  = scale + NROWS;                             // N

    const dim3 blk(256);
    const dim3 gemmGrid(NROWS / BM);          // 128 workgroups
    const size_t Asz = (size_t)NROWS * NROWS;

    // ---- bottom_to_up level 0: emb = (emb + A0 @ emb) * scale0
    rowscale_kernel<<<NROWS, blk, 0, stream>>>(d2u, scale);
    gemm_wmma_f32<0><<<gemmGrid, blk, 0, stream>>>(d2u, embedding, embedding, scale,
                                                   buf0, NROWS);
    // ---- level 1
    rowscale_kernel<<<NROWS, blk, 0, stream>>>(d2u + Asz, scale);
    gemm_wmma_f32<0><<<gemmGrid, blk, 0, stream>>>(d2u + Asz, buf0, buf0, scale,
                                                   buf1, NROWS);

    // ---- GCNConv 1 (same-level): input buf1 -> output buf0
    deg_init_kernel  <<<NROWS / 256, blk, 0, stream>>>(dinv);
    deg_count_kernel <<<NEDGE / 256, blk, 0, stream>>>(same_ei + NEDGE, dinv);
    deg_finish_kernel<<<NROWS / 256, blk, 0, stream>>>(dinv);
    gemm_wmma_f32<1><<<gemmGrid, blk, 0, stream>>>(buf1, W_same, nullptr, nullptr,
                                                   bufh, DF);
    selfloop_bias_kernel<<<(NROWS * DF) / 256, blk, 0, stream>>>(bufh, dinv, b_same, buf0);
    scatter_edges_kernel<<<NEDGE / 4, blk, 0, stream>>>(same_ei, same_ei + NEDGE,
                                                        dinv, bufh, buf0);

    // ---- GCNConv 2 (top-to-bottom): input buf0 -> output d_out
    deg_init_kernel  <<<NROWS / 256, blk, 0, stream>>>(dinv);
    deg_count_kernel <<<NEDGE / 256, blk, 0, stream>>>(u2d_ei + NEDGE, dinv);
    deg_finish_kernel<<<NROWS / 256, blk, 0, stream>>>(dinv);
    gemm_wmma_f32<1><<<gemmGrid, blk, 0, stream>>>(buf0, W_top, nullptr, nullptr,
                                                   bufh, DF);
    selfloop_bias_kernel<<<(NROWS * DF) / 256, blk, 0, stream>>>(bufh, dinv, b_top, out);
    scatter_edges_kernel<<<NEDGE / 4, blk, 0, stream>>>(u2d_ei, u2d_ei + NEDGE,
                                                        dinv, bufh, out);
}